// Net_79465484911206
// MI455X (gfx1250) — compile-verified
//
#include <hip/hip_runtime.h>

typedef float v2f __attribute__((ext_vector_type(2)));
typedef float v4f __attribute__((ext_vector_type(4)));
typedef float v8f __attribute__((ext_vector_type(8)));

#define TT 200
#define FF 64
#define NW 256   // batch tiles: 4096 / 16

// -------------------------------------------------------------------------
// Kernel 1: x_proj[t][b][g] = x[b][t][:] . w_ih[g][:] + b_ih[g]
// Computed transposed: D[g][m] = sum_k w_ih[g][k] * x[b0+m][t][k]
//   A (16x4 f32, per WMMA): lane l -> row M=l&15, VGPR0=K=k0+2*(l>>4), VGPR1=+1
//   B (4x16 f32):           lane l -> col N=l&15, VGPR0=K=2*(l>>4),   VGPR1=+1
//   D (16x16 f32):          VGPR j, lane l -> row j+8*(l>>4), col l&15
// x tile staged with GLOBAL_LOAD_ASYNC_TO_LDS_B128 (ASYNCcnt), non-temporal
// so the 210MB x stream does not evict the L2-resident x_proj intermediate.
// Output layout per (t, wid): 12 words x 32 lanes:
//   word j   (0..7): xp[b0+(l&15)][ j      + 8*(l>>4)]   (r | z halves)
//   word 8+j (0..3): xp[b0+(l&15)][ 16 + j + 4*(l>>4)]   (n gate, packed)
// -------------------------------------------------------------------------
__global__ __launch_bounds__(256)
void gru_proj(const float* __restrict__ x, const float* __restrict__ w_ih,
              const float* __restrict__ b_ih, float* __restrict__ XP, int t0)
{
    __shared__ __align__(16) float wlds[32 * 68];    // w_ih, 32 rows, stride 68
    __shared__ __align__(16) float xlds[8][16 * 68]; // per-wave x tile, stride 68

    const int tid  = threadIdx.x;
    const int wave = tid >> 5;
    const int lane = tid & 31;

    for (int i = tid; i < 32 * 68; i += 256) wlds[i] = 0.0f;
    __syncthreads();
    for (int i = tid; i < 24 * 64; i += 256) {
        int g = i >> 6, k = i & 63;
        wlds[g * 68 + k] = w_ih[i];
    }
    __syncthreads();

    const int tile = blockIdx.x * 8 + wave;
    const int wid  = tile & (NW - 1);
    const int lt   = tile >> 8;
    const int t    = t0 + lt;
    const int b0   = wid * 16;

    // async-stage 16x64 f32 x tile into LDS: per lane 8 x b128, global and
    // LDS addresses advance by the same 16B within a 32-col half row.
    {
        const int row = lane >> 1, halfc = (lane & 1) * 32;
        const float* src = x + ((long)(b0 + row) * TT + t) * FF + halfc;
        unsigned ldsoff = (unsigned)(uintptr_t)&xlds[wave][row * 68 + halfc];
        #pragma unroll
        for (int c = 0; c < 8; ++c) {
            unsigned long long ga = (unsigned long long)(uintptr_t)(src + c * 4);
            unsigned la = ldsoff + c * 16;
            asm volatile("global_load_async_to_lds_b128 %0, %1, off th:TH_LOAD_NT"
                         :: "v"(la), "v"(ga) : "memory");
        }
    }

    const int m = lane & 15, hsel = lane >> 4;

    v8f acc0, acc1;                        // C init = bias broadcast
    #pragma unroll
    for (int j = 0; j < 8; ++j) {
        acc0[j] = b_ih[j + 8 * hsel];
        int g1 = 16 + j + 8 * hsel;
        acc1[j] = (g1 < 24) ? b_ih[g1] : 0.0f;
    }

    asm volatile("s_wait_asynccnt 0x0" ::: "memory");

    const float* xs = &xlds[wave][0];
    #pragma unroll
    for (int i = 0; i < 16; ++i) {         // K = 64 in 16 chunks of 4
        const int k = 4 * i + 2 * hsel;
        v2f B  = *(const v2f*)(xs   + m * 68 + k);
        v2f A0 = *(const v2f*)(wlds + m * 68 + k);
        v2f A1 = *(const v2f*)(wlds + (m + 16) * 68 + k);
        acc0 = __builtin_amdgcn_wmma_f32_16x16x4_f32(false, A0, false, B,
                                                     (short)0, acc0, false, false);
        acc1 = __builtin_amdgcn_wmma_f32_16x16x4_f32(false, A1, false, B,
                                                     (short)0, acc1, false, false);
    }

    float* out = XP + ((size_t)lt * NW + wid) * 384 + lane;
    #pragma unroll
    for (int j = 0; j < 8; ++j) out[j * 32] = acc0[j];
    #pragma unroll
    for (int j = 0; j < 4; ++j) {          // pack n-gate: hi half from VGPR j+4
        float hi = __shfl_xor(acc1[j + 4], 16);
        out[(8 + j) * 32] = (lane < 16) ? acc1[j] : hi;
    }
}

// -------------------------------------------------------------------------
// Kernel 2: sequential GRU scan, one wave per 16 batch rows.
// gh = w_hh . h^T + b_hh via 4 chained f32 WMMAs per step.
// h kept lane-replicated: h[j] lane l = h[b0+(l&15)][j] (both halves).
// Gate math packed full-width: n/z/h-update work on (j | j+4) lane halves.
// -------------------------------------------------------------------------
__global__ __launch_bounds__(32)
void gru_scan(const float* __restrict__ XP, const float* __restrict__ w_hh,
              const float* __restrict__ b_hh, const float* __restrict__ w_dec,
              const float* __restrict__ b_dec, float* __restrict__ HWS,
              float* __restrict__ out, int ct, int is_first, int is_last)
{
    const int wid  = blockIdx.x;
    const int lane = threadIdx.x;
    const int m = lane & 15, hsel = lane >> 4;
    const bool lo = (lane < 16);

    v2f A[2][2];                           // w_hh WMMA A operands (constant)
    #pragma unroll
    for (int mt = 0; mt < 2; ++mt)
        #pragma unroll
        for (int c = 0; c < 2; ++c) {
            int g = 16 * mt + m;
            int k = 4 * c + 2 * hsel;
            if (g < 24) A[mt][c] = *(const v2f*)(w_hh + g * 8 + k);
            else        A[mt][c] = (v2f){0.0f, 0.0f};
        }

    v8f cb0, cb1;                          // b_hh broadcast in D layout
    #pragma unroll
    for (int j = 0; j < 8; ++j) {
        cb0[j] = b_hh[j + 8 * hsel];
        int g1 = 16 + j + 8 * hsel;
        cb1[j] = (g1 < 24) ? b_hh[g1] : 0.0f;
    }

    float h[8];
    #pragma unroll
    for (int j = 0; j < 8; ++j)
        h[j] = is_first ? 0.0f : HWS[wid * 256 + j * 32 + lane];

    #pragma unroll 2
    for (int lt = 0; lt < ct; ++lt) {
        const float* xp = XP + ((size_t)lt * NW + wid) * 384 + lane;
        float xrz[8], xn4[4];
        #pragma unroll
        for (int j = 0; j < 8; ++j) xrz[j] = xp[j * 32];
        #pragma unroll
        for (int j = 0; j < 4; ++j) xn4[j] = xp[(8 + j) * 32];

        // B operands from replicated h: pure cndmask, no cross-lane
        v2f B0 = { lo ? h[0] : h[2], lo ? h[1] : h[3] };
        v2f B1 = { lo ? h[4] : h[6], lo ? h[5] : h[7] };

        v8f d0 = __builtin_amdgcn_wmma_f32_16x16x4_f32(false, A[0][0], false, B0,
                                                       (short)0, cb0, false, false);
        d0     = __builtin_amdgcn_wmma_f32_16x16x4_f32(false, A[0][1], false, B1,
                                                       (short)0, d0, false, false);
        v8f d1 = __builtin_amdgcn_wmma_f32_16x16x4_f32(false, A[1][0], false, B0,
                                                       (short)0, cb1, false, false);
        d1     = __builtin_amdgcn_wmma_f32_16x16x4_f32(false, A[1][1], false, B1,
                                                       (short)0, d1, false, false);

        // sigmoid on full width: lanes<16 -> r_j, lanes>=16 -> z_j
        float s[8];
        #pragma unroll
        for (int j = 0; j < 8; ++j) {
            float pre = d0[j] + xrz[j];
            s[j] = __builtin_amdgcn_rcpf(1.0f + __expf(-pre));
        }

        // packed gate update: reg j' carries gate j' (lo half) and j'+4 (hi)
        #pragma unroll
        for (int j = 0; j < 4; ++j) {
            float rp = lo ? s[j]  : __shfl_xor(s[j + 4], 16);   // r packed
            float gp = lo ? d1[j] : __shfl_xor(d1[j + 4], 16);  // ghn packed
            float np = xn4[j] + rp * gp;
            float e  = __expf(2.0f * np);                       // tanh, robust
            float n  = 1.0f - 2.0f * __builtin_amdgcn_rcpf(e + 1.0f);
            float zp = lo ? __shfl_xor(s[j], 16) : s[j + 4];    // z packed
            float hp = lo ? h[j] : h[j + 4];
            float hn = (1.0f - zp) * n + zp * hp;               // full width
            float hs = __shfl_xor(hn, 16);
            h[j]     = lo ? hn : hs;                            // re-replicate
            h[j + 4] = lo ? hs : hn;
        }
    }

    #pragma unroll
    for (int j = 0; j < 8; ++j) HWS[wid * 256 + j * 32 + lane] = h[j];

    if (is_last && lane < 16) {
        float acc = b_dec[0];
        #pragma unroll
        for (int j = 0; j < 8; ++j) acc += h[j] * w_dec[j];
        out[wid * 16 + lane] = acc;
    }
}

// -------------------------------------------------------------------------
extern "C" void kernel_launch(void* const* d_in, const int* in_sizes, int n_in,
                              void* d_out, int out_size, void* d_ws, size_t ws_size,
                              hipStream_t stream)
{
    const float* x     = (const float*)d_in[0];
    const float* w_ih  = (const float*)d_in[1];
    const float* w_hh  = (const float*)d_in[2];
    const float* b_ih  = (const float*)d_in[3];
    const float* b_hh  = (const float*)d_in[4];
    const float* w_dec = (const float*)d_in[5];
    const float* b_dec = (const float*)d_in[6];
    float* out = (float*)d_out;

    float* HWS = (float*)d_ws;                       // 256 KB h carry state
    float* XP  = HWS + NW * 8 * 32;                  // x_proj staging
    const size_t hws_bytes = (size_t)NW * 8 * 32 * 4;
    const size_t per_t     = (size_t)NW * 12 * 32 * 4;   // 384 KB / timestep

    long ctmax = (ws_size > hws_bytes) ? (long)((ws_size - hws_bytes) / per_t) : 1;
    int chunk = (int)(ctmax < 1 ? 1 : (ctmax > TT ? TT : ctmax));

    for (int t0 = 0; t0 < TT; t0 += chunk) {
        int ct = (TT - t0 < chunk) ? (TT - t0) : chunk;
        gru_proj<<<(NW * ct) / 8, 256, 0, stream>>>(x, w_ih, b_ih, XP, t0);
        gru_scan<<<NW, 32, 0, stream>>>(XP, w_hh, b_hh, w_dec, b_dec, HWS, out,
                                        ct, t0 == 0, t0 + ct >= TT);
    }
}